// DFFormerMoELoss_54348516163744
// MI455X (gfx1250) — compile-verified
//
#include <hip/hip_runtime.h>
#include <hip/hip_bf16.h>

#define BN 16
#define NN 2048

typedef __attribute__((ext_vector_type(2))) float v2f;
typedef __attribute__((ext_vector_type(8))) float v8f;

#define LOG2E_F 1.44269504088896340736f
#define LN2_F   0.69314718055994530942f

// ws layout (floats): [0..15] pair loss sums, [16..31] pair pos counts, [32..47] listmle ll per row
__global__ __launch_bounds__(64) void init_ws_kernel(float* ws) {
    if (threadIdx.x < 48) ws[threadIdx.x] = 0.0f;
}

__device__ __forceinline__ float softplus_f(float x) {
    // softplus(x) = max(x,0) + ln2 * log2(1 + 2^(-|x|*log2e))
    // Raw v_exp_f32 / v_log_f32; arguments are in the well-conditioned range.
    const float e = __builtin_amdgcn_exp2f(-fabsf(x) * LOG2E_F);   // in (0, 1]
    return fmaf(LN2_F, __builtin_amdgcn_logf(1.0f + e), fmaxf(x, 0.0f));
}

// Pairwise logistic: grid = BN*8 blocks, 256 threads (8 waves).
// Each block: one row b, one 256-wide i segment; j sweeps full row from LDS.
// Each wave accumulates 4(i)x16(j) tiles via V_WMMA_F32_16X16X4_F32 with A=ones.
__global__ __launch_bounds__(256) void pairwise_kernel(const float* __restrict__ pred,
                                                       const float* __restrict__ tru,
                                                       const float* __restrict__ masks,
                                                       float* __restrict__ ws) {
    __shared__ float smp[NN];
    __shared__ float smt[NN];
    const int b   = blockIdx.x >> 3;
    const int seg = blockIdx.x & 7;
    const int t   = threadIdx.x;

    const float* pr = pred  + (size_t)b * NN;
    const float* tr = tru   + (size_t)b * NN;
    const float* mk = masks + (size_t)b * NN;
    __builtin_prefetch(pr, 0, 1);   // global_prefetch_b8
    __builtin_prefetch(tr, 0, 1);

    for (int idx = t; idx < NN; idx += 256) {
        float m = mk[idx];
        smp[idx] = pr[idx] * m;
        smt[idx] = tr[idx] * m;
    }
    __syncthreads();

    const int wave = t >> 5;
    const int lane = t & 31;
    const int n    = lane & 15;   // j column within tile
    const int half = lane >> 4;   // k-half: lanes 0-15 -> k={0,1}, lanes 16-31 -> k={2,3}

    v8f acc = {};   // softplus sums (D tile, all rows identical)
    v8f cnt = {};   // pos counts
    v2f ones; ones.x = 1.0f; ones.y = 1.0f;   // A = ones 16x4

    const int ibase0 = seg * 256 + wave * 32;   // 32 i's per wave
    #pragma unroll 1
    for (int it = 0; it < 8; ++it) {
        const int ib = ibase0 + it * 4;
        const int i0 = ib + half * 2;
        const float pi0 = smp[i0],     ti0 = smt[i0];
        const float pi1 = smp[i0 + 1], ti1 = smt[i0 + 1];
        #pragma unroll 4
        for (int jb = 0; jb < NN; jb += 16) {
            const float pj = smp[jb + n];
            const float tj = smt[jb + n];

            const float td0 = ti0 - tj;
            const float td1 = ti1 - tj;
            const float sp0 = softplus_f(pj - pi0);   // softplus(-margin*pd), margin=1
            const float sp1 = softplus_f(pj - pi1);

            v2f bv, cv;   // B matrix 4x16 f32 layout: 2 VGPRs, lane half selects k pair
            bv.x = (td0 > 0.0f) ? sp0 : 0.0f;
            bv.y = (td1 > 0.0f) ? sp1 : 0.0f;
            cv.x = (td0 > 0.0f) ? 1.0f : 0.0f;
            cv.y = (td1 > 0.0f) ? 1.0f : 0.0f;

            acc = __builtin_amdgcn_wmma_f32_16x16x4_f32(false, ones, false, bv,
                                                        (short)0, acc, false, false);
            cnt = __builtin_amdgcn_wmma_f32_16x16x4_f32(false, ones, false, cv,
                                                        (short)0, cnt, false, false);
        }
    }

    // D rows are identical; VGPR0 carries (M=0,N=lane) for lanes 0-15 and (M=8,N) for 16-31,
    // so a full 32-lane sum is exactly 2x the tile's column-sum total.
    float a = acc[0];
    float c = cnt[0];
    #pragma unroll
    for (int off = 16; off > 0; off >>= 1) {
        a += __shfl_xor(a, off);
        c += __shfl_xor(c, off);
    }
    if (lane == 0) {
        atomicAdd(&ws[b],      0.5f * a);
        atomicAdd(&ws[16 + b], 0.5f * c);
    }
}

__device__ __forceinline__ float lse2(float a, float b) {
    const float e = __builtin_amdgcn_exp2f(-fabsf(a - b) * LOG2E_F);
    return fmaf(LN2_F, __builtin_amdgcn_logf(1.0f + e), fmaxf(a, b));
}

// ListMLE: one 1024-thread block per row. Bitonic sort (desc by masked true) of
// (key,val) in LDS, reverse inclusive logsumexp scan, then sum(sp - lse)/N.
__global__ __launch_bounds__(1024) void listmle_kernel(const float* __restrict__ pred,
                                                       const float* __restrict__ tru,
                                                       const float* __restrict__ masks,
                                                       float* __restrict__ ws) {
    __shared__ float skey[NN];
    __shared__ float sval[NN];
    __shared__ float sred[32];
    const int b = blockIdx.x;
    const int t = threadIdx.x;

    for (int idx = t; idx < NN; idx += 1024) {
        float m = masks[(size_t)b * NN + idx];
        skey[idx] = tru[(size_t)b * NN + idx] * m;
        sval[idx] = pred[(size_t)b * NN + idx] * m;
    }
    __syncthreads();

    // Bitonic sort, descending by key.
    for (int size = 2; size <= NN; size <<= 1) {
        for (int stride = size >> 1; stride > 0; stride >>= 1) {
            const int pos = 2 * t - (t & (stride - 1));
            const int q   = pos + stride;
            const bool dirAsc = ((pos & size) != 0);   // inverted -> overall descending
            const float kp = skey[pos], kq = skey[q];
            const bool sw = dirAsc ? (kp > kq) : (kp < kq);
            if (sw) {
                skey[pos] = kq; skey[q] = kp;
                const float vp = sval[pos], vq = sval[q];
                sval[pos] = vq; sval[q] = vp;
            }
            __syncthreads();
        }
    }

    // Reverse inclusive logsumexp scan (Hillis-Steele); reuse skey as the scan array.
    for (int idx = t; idx < NN; idx += 1024) skey[idx] = sval[idx];
    __syncthreads();
    for (int d = 1; d < NN; d <<= 1) {
        const int j0 = t, j1 = t + 1024;
        float r0 = 0.0f, r1 = 0.0f;
        const bool w0 = (j0 + d) < NN;
        const bool w1 = (j1 + d) < NN;
        if (w0) r0 = lse2(skey[j0], skey[j0 + d]);
        if (w1) r1 = lse2(skey[j1], skey[j1 + d]);
        __syncthreads();
        if (w0) skey[j0] = r0;
        if (w1) skey[j1] = r1;
        __syncthreads();
    }

    // sum_j (sp_j - lse_j)
    float s = (sval[t] - skey[t]) + (sval[t + 1024] - skey[t + 1024]);
    const int lane = t & 31, wv = t >> 5;
    #pragma unroll
    for (int off = 16; off > 0; off >>= 1) s += __shfl_xor(s, off);
    if (lane == 0) sred[wv] = s;
    __syncthreads();
    if (wv == 0) {
        float v = sred[lane];
        #pragma unroll
        for (int off = 16; off > 0; off >>= 1) v += __shfl_xor(v, off);
        if (lane == 0) ws[32 + b] = v * (1.0f / (float)NN);
    }
}

__global__ __launch_bounds__(32) void finalize_kernel(const float* __restrict__ ws,
                                                      const float* __restrict__ lb,
                                                      float* __restrict__ out) {
    if (threadIdx.x != 0) return;
    float sll = 0.0f;
    for (int b = 0; b < BN; ++b) sll += ws[32 + b];
    const float ranking = -sll / (float)BN;
    float pw = 0.0f;
    for (int b = 0; b < BN; ++b) pw += ws[b] / (ws[16 + b] + 1e-12f);
    pw /= (float)BN;
    out[0] = 1.0f * ranking + 0.3f * pw + 0.03f * lb[0];
}

extern "C" void kernel_launch(void* const* d_in, const int* in_sizes, int n_in,
                              void* d_out, int out_size, void* d_ws, size_t ws_size,
                              hipStream_t stream) {
    const float* y_pred = (const float*)d_in[0];
    const float* y_true = (const float*)d_in[1];
    const float* masks  = (const float*)d_in[2];
    const float* lb     = (const float*)d_in[3];
    float* out = (float*)d_out;
    float* ws  = (float*)d_ws;

    init_ws_kernel<<<1, 64, 0, stream>>>(ws);
    pairwise_kernel<<<BN * 8, 256, 0, stream>>>(y_pred, y_true, masks, ws);
    listmle_kernel<<<BN, 1024, 0, stream>>>(y_pred, y_true, masks, ws);
    finalize_kernel<<<1, 32, 0, stream>>>(ws, lb, out);
}